// MambaConv1D_17136919511393
// MI455X (gfx1250) — compile-verified
//
#include <hip/hip_runtime.h>

// Depthwise causal conv1d (K=4, pad=3), NCL layout.
// x: (B,C,L) f32, kernel: (K,1,C) f32, bias: (C) f32, out: (B,C,L+3) f32
// y[b,c,t] = sum_{k=0..3} w[k,c] * x[b,c, t-3+k] + bias[c],  x out of range -> 0

constexpr int Cc   = 4096;
constexpr int Lc   = 4096;
constexpr int LOUT = 4099;   // L + 2*3 - (4-1)

typedef float v4f __attribute__((ext_vector_type(4)));  // native vector: OK for nontemporal builtin

__global__ __launch_bounds__(256) void mamba_dwconv1d(
    const float* __restrict__ x,
    const float* __restrict__ kern,   // (K,1,C)
    const float* __restrict__ bias,   // (C,)
    float* __restrict__ out)          // (B,C,LOUT)
{
    // layout: smem[1..3] = x[-3..-1] = 0 ; smem[4+j] = x[j] ; smem[4100..4102] = x[4096..4098] = 0
    __shared__ float smem[4104];

    const int tid = threadIdx.x;
    const int row = blockIdx.x;        // b*C + c
    const int c   = row & (Cc - 1);

    // zero front halo smem[0..3] and tail halo smem[4100..4103]
    if (tid < 4) {
        smem[tid]        = 0.0f;
        smem[4100 + tid] = 0.0f;
    }

    // Async-copy the 16 KB x row into LDS (CDNA5 async global->LDS path, ASYNCcnt).
    // 1024 x 16-byte chunks; each of 256 lanes moves 4 chunks; 16B-aligned LDS dest.
    const float* gx = x + (size_t)row * Lc;
    #pragma unroll
    for (int i = 0; i < 4; ++i) {
        const int chunk = tid + i * 256;                             // 0..1023
        const float* gaddr = gx + chunk * 4;                         // 16B granularity
        unsigned ldsb = (unsigned)(uintptr_t)(smem + 4 + chunk * 4); // LDS byte addr
        asm volatile("global_load_async_to_lds_b128 %0, %1, off"
                     :: "v"(ldsb), "v"(gaddr) : "memory");
    }
    // wait for this wave's async copies, then make all waves' copies visible
    asm volatile("s_wait_asynccnt 0" ::: "memory");
    __syncthreads();

    // per-row taps + bias are uniform across the block -> scalar (SGPR) loads
    const float w0 = kern[0 * Cc + c];
    const float w1 = kern[1 * Cc + c];
    const float w2 = kern[2 * Cc + c];
    const float w3 = kern[3 * Cc + c];
    const float bz = bias[c];

    float* gy = out + (size_t)row * LOUT;

    // Row base (row*4099 dwords) is 3*row (mod 4) dwords off 16B alignment;
    // phase a = row & 3 makes (row*4099 + a + 4j) % 4 == 0, so every v4f
    // store below is 16B-aligned. Row = a head dwords + 1024 float4 chunks
    // + (3 - a) tail dwords.
    const int a = row & 3;

    #pragma unroll
    for (int k = 0; k < 4; ++k) {
        const int j  = tid + k * 256;        // chunk id 0..1023
        const int t0 = a + 4 * j;            // first output of this chunk
        // taps for outputs t0..t0+3: x[t0-3 .. t0+3] == smem[t0+1 .. t0+7]
        const float v0 = smem[t0 + 1];
        const float v1 = smem[t0 + 2];
        const float v2 = smem[t0 + 3];
        const float v3 = smem[t0 + 4];
        const float v4 = smem[t0 + 5];
        const float v5 = smem[t0 + 6];
        const float v6 = smem[t0 + 7];
        v4f r;
        r.x = fmaf(w3, v3, fmaf(w2, v2, fmaf(w1, v1, fmaf(w0, v0, bz))));
        r.y = fmaf(w3, v4, fmaf(w2, v3, fmaf(w1, v2, fmaf(w0, v1, bz))));
        r.z = fmaf(w3, v5, fmaf(w2, v4, fmaf(w1, v3, fmaf(w0, v2, bz))));
        r.w = fmaf(w3, v6, fmaf(w2, v5, fmaf(w1, v4, fmaf(w0, v3, bz))));
        // output is touch-once: non-temporal 128-bit store (aligned by phase a)
        __builtin_nontemporal_store(r, reinterpret_cast<v4f*>(gy + t0));
    }

    // head outputs t = 0 .. a-1 (front halo taps are zeroed)
    if (tid < a) {
        const int t = tid;
        float s = bz;
        s = fmaf(w0, smem[t + 1], s);
        s = fmaf(w1, smem[t + 2], s);
        s = fmaf(w2, smem[t + 3], s);
        s = fmaf(w3, smem[t + 4], s);
        __builtin_nontemporal_store(s, gy + t);
    }
    // tail outputs t = a+4096 .. 4098 (taps run into zeroed tail halo)
    if (tid < 3 - a) {
        const int t = a + Lc + tid;
        float s = bz;
        s = fmaf(w0, smem[t + 1], s);
        s = fmaf(w1, smem[t + 2], s);
        s = fmaf(w2, smem[t + 3], s);
        s = fmaf(w3, smem[t + 4], s);
        __builtin_nontemporal_store(s, gy + t);
    }
}

extern "C" void kernel_launch(void* const* d_in, const int* in_sizes, int n_in,
                              void* d_out, int out_size, void* d_ws, size_t ws_size,
                              hipStream_t stream) {
    const float* x    = (const float*)d_in[0];   // (4,4096,4096)
    const float* kern = (const float*)d_in[1];   // (4,1,4096)
    const float* bias = (const float*)d_in[2];   // (4096,)
    float* out = (float*)d_out;                  // (4,4096,4099)

    const int rows = 4 * Cc;                     // B*C = 16384 blocks
    mamba_dwconv1d<<<rows, 256, 0, stream>>>(x, kern, bias, out);
}